// MWERLoss_3513283248530
// MI455X (gfx1250) — compile-verified
//
#include <hip/hip_runtime.h>
#include <math.h>

#define B_ 16
#define T_ 1024
#define V_ 4096
#define P_ 100
#define L_ 128
#define BIGI (1 << 28)

typedef __attribute__((ext_vector_type(2))) float v2f;
typedef __attribute__((ext_vector_type(8))) float v8f;

__device__ __forceinline__ float wave_max(float v) {
#pragma unroll
    for (int off = 16; off >= 1; off >>= 1) v = fmaxf(v, __shfl_xor(v, off, 32));
    return v;
}
__device__ __forceinline__ float wave_sum(float v) {
#pragma unroll
    for (int off = 16; off >= 1; off >>= 1) v += __shfl_xor(v, off, 32);
    return v;
}

// ---------------------------------------------------------------------------
// Kernel 1: lse[b,t] = logsumexp_v emissions[b,t,v].  One 256-thread block per
// row; row (16KB) cached in registers; exp-sum reduced per-wave with a single
// V_WMMA_F32_16X16X4_F32 against an all-ones B matrix (f32 accumulate).
// ---------------------------------------------------------------------------
__global__ __launch_bounds__(256) void lse_kernel(const float* __restrict__ emis,
                                                  float* __restrict__ lse) {
    const int row  = blockIdx.x;           // b*T + t
    const int tid  = threadIdx.x;
    const int lane = tid & 31;
    const int wid  = tid >> 5;

    const float4* rp = (const float4*)(emis + (size_t)row * V_);
    float4 x0 = rp[tid];
    float4 x1 = rp[tid + 256];
    float4 x2 = rp[tid + 512];
    float4 x3 = rp[tid + 768];

    float m = fmaxf(fmaxf(x0.x, x0.y), fmaxf(x0.z, x0.w));
    m = fmaxf(m, fmaxf(fmaxf(x1.x, x1.y), fmaxf(x1.z, x1.w)));
    m = fmaxf(m, fmaxf(fmaxf(x2.x, x2.y), fmaxf(x2.z, x2.w)));
    m = fmaxf(m, fmaxf(fmaxf(x3.x, x3.y), fmaxf(x3.z, x3.w)));
    m = wave_max(m);

    __shared__ float smax[8];
    __shared__ float ssum[8];
    if (lane == 0) smax[wid] = m;
    __syncthreads();
    float rmax = smax[0];
#pragma unroll
    for (int i = 1; i < 8; i++) rmax = fmaxf(rmax, smax[i]);

    // two partial exp-sums per lane -> A-matrix operand (16x4 f32)
    float pa = __expf(x0.x - rmax) + __expf(x0.y - rmax) +
               __expf(x0.z - rmax) + __expf(x0.w - rmax) +
               __expf(x1.x - rmax) + __expf(x1.y - rmax) +
               __expf(x1.z - rmax) + __expf(x1.w - rmax);
    float pb = __expf(x2.x - rmax) + __expf(x2.y - rmax) +
               __expf(x2.z - rmax) + __expf(x2.w - rmax) +
               __expf(x3.x - rmax) + __expf(x3.y - rmax) +
               __expf(x3.z - rmax) + __expf(x3.w - rmax);

    // D = A x Ones : D[m][n] = sum_k A[m][k]; every (lane,vgpr) slot of A maps
    // to a distinct (m,k), so the 64 partials are reduced to 16 row-sums in
    // one matrix op, f32 all the way. EXEC is all ones here (no divergence).
    v2f a    = {pa, pb};
    v2f ones = {1.0f, 1.0f};
    v8f c    = {};
    c = __builtin_amdgcn_wmma_f32_16x16x4_f32(false, a, false, ones,
                                              (short)0, c, false, false);
    float h = c[0] + c[1] + c[2] + c[3] + c[4] + c[5] + c[6] + c[7];
    h += __shfl_xor(h, 16, 32);   // rows 0-7 half + rows 8-15 half

    if (lane == 0) ssum[wid] = h;
    __syncthreads();
    if (tid == 0) {
        float s = ssum[0];
#pragma unroll
        for (int i = 1; i < 8; i++) s += ssum[i];
        lse[row] = rmax + __logf(s);
    }
}

// ---------------------------------------------------------------------------
// Kernel 2: scores[b,p] = sum_{t<len_b} emissions[b,t,paths[b,p,t]] - lse[b,t]
// One wave32 per (b,p); coalesced token loads, gathered emission reads.
// ---------------------------------------------------------------------------
__global__ __launch_bounds__(256) void score_kernel(const float* __restrict__ emis,
                                                    const int* __restrict__ elens,
                                                    const int* __restrict__ paths,
                                                    const float* __restrict__ lse,
                                                    float* __restrict__ scores) {
    const int w    = blockIdx.x * 8 + (threadIdx.x >> 5);   // 0..B*P-1
    const int lane = threadIdx.x & 31;
    const int b    = w / P_;
    const int p    = w % P_;
    const int len  = elens[b];

    const int*   pr = paths + (size_t)(b * P_ + p) * T_;
    const float* eb = emis + (size_t)b * T_ * V_;
    const float* lb = lse + b * T_;

    float acc = 0.0f;
    for (int t0 = 0; t0 < T_; t0 += 32) {
        if (t0 >= len) break;                 // uniform early exit
        if (t0 + 64 < T_) __builtin_prefetch(pr + t0 + 64, 0, 0);
        int t = t0 + lane;
        if (t < len) {
            int tok = pr[t];
            acc += eb[(size_t)t * V_ + tok] - lb[t];
        }
    }
    acc = wave_sum(acc);
    if (lane == 0) scores[w] = acc;
}

// ---------------------------------------------------------------------------
// Kernel 3: Levenshtein(CTC-collapse(path), labels) per (b,p).
// One wave32 per (b,p); lane i owns DP columns [5i, 5i+4]; horizontal
// dependency resolved with the prefix-min-of-(cand - j) scan (as reference).
// Blank / repeat / out-of-length steps skipped with a wave-uniform branch.
// ---------------------------------------------------------------------------
__global__ __launch_bounds__(256) void edit_kernel(const int* __restrict__ paths,
                                                   const int* __restrict__ elens,
                                                   const int* __restrict__ labels,
                                                   const int* __restrict__ llens,
                                                   float* __restrict__ wers) {
    __shared__ int slab[L_];
    const int blocks_per_b = (P_ + 7) / 8;              // 13
    const int b     = blockIdx.x / blocks_per_b;
    const int pbase = (blockIdx.x % blocks_per_b) * 8;
    const int wid   = threadIdx.x >> 5;
    const int lane  = threadIdx.x & 31;

    for (int i = threadIdx.x; i < L_; i += 256) slab[i] = labels[b * L_ + i];
    __syncthreads();

    const int p = pbase + wid;
    if (p >= P_) return;

    const int len   = elens[b];
    const int llen  = llens[b];
    const int jbase = lane * 5;

    // reference labels for columns j = jbase..jbase+4  (sub uses labels[j-1])
    int r0 = (jbase - 1 >= 0 && jbase - 1 < L_) ? slab[jbase - 1] : -2;
    int r1 = (jbase     < L_) ? slab[jbase]     : -2;
    int r2 = (jbase + 1 < L_) ? slab[jbase + 1] : -2;
    int r3 = (jbase + 2 < L_) ? slab[jbase + 2] : -2;
    int r4 = (jbase + 3 < L_) ? slab[jbase + 3] : -2;

    int D0 = (jbase     <= L_) ? jbase     : BIGI;
    int D1 = (jbase + 1 <= L_) ? jbase + 1 : BIGI;
    int D2 = (jbase + 2 <= L_) ? jbase + 2 : BIGI;
    int D3 = (jbase + 3 <= L_) ? jbase + 3 : BIGI;
    int D4 = (jbase + 4 <= L_) ? jbase + 4 : BIGI;

    const int* pr = paths + (size_t)(b * P_ + p) * T_;
    int carry = -1;   // token before t=0

    for (int t0 = 0; t0 < T_; t0 += 32) {
        if (t0 >= len) break;                        // uniform
        int tok = pr[t0 + lane];
        int lim = min(32, len - t0);
        for (int s = 0; s < lim; s++) {
            int c  = __shfl(tok, s, 32);
            int pc = (s == 0) ? carry : __shfl(tok, s - 1, 32);
            if (c != 0 && c != pc) {                 // uniform valid-step test
                int pd = __shfl_up(D4, 1, 32);       // D[jbase-1] from lane-1
                int c0 = D0 + 1;
                if (lane) c0 = min(c0, pd + (r0 != c));
                int c1 = min(D1 + 1, D0 + (r1 != c));
                int c2 = min(D2 + 1, D1 + (r2 != c));
                int c3 = min(D3 + 1, D2 + (r3 != c));
                int c4 = min(D4 + 1, D3 + (r4 != c));
                int m0 = c0 - jbase;
                int m1 = min(m0, c1 - (jbase + 1));
                int m2 = min(m1, c2 - (jbase + 2));
                int m3 = min(m2, c3 - (jbase + 3));
                int m4 = min(m3, c4 - (jbase + 4));
                // wave-level exclusive prefix-min of lane totals
                int sc = m4;
#pragma unroll
                for (int off = 1; off < 32; off <<= 1) {
                    int u = __shfl_up(sc, off, 32);
                    if (lane >= off) sc = min(sc, u);
                }
                int ex = __shfl_up(sc, 1, 32);
                if (lane == 0) ex = BIGI;
                D0 = min(m0, ex) + jbase;
                D1 = min(m1, ex) + jbase + 1;
                D2 = min(m2, ex) + jbase + 2;
                D3 = min(m3, ex) + jbase + 3;
                D4 = min(m4, ex) + jbase + 4;
                if (jbase + 4 > L_) {                // keep invalid cols inert
                    if (jbase     > L_) D0 = BIGI;
                    if (jbase + 1 > L_) D1 = BIGI;
                    if (jbase + 2 > L_) D2 = BIGI;
                    if (jbase + 3 > L_) D3 = BIGI;
                    D4 = BIGI;
                }
            }
        }
        carry = __shfl(tok, 31, 32);
    }

    const int q   = llen % 5;
    int sel = (q == 0) ? D0 : (q == 1) ? D1 : (q == 2) ? D2 : (q == 3) ? D3 : D4;
    int wer = __shfl(sel, llen / 5, 32);
    if (lane == 0) wers[b * P_ + p] = (float)wer;
}

// ---------------------------------------------------------------------------
// Kernel 4: out = sum_b sum_p softmax_p(scores[b,:]) * wers[b,p]
// Single block, deterministic summation order.
// ---------------------------------------------------------------------------
__global__ __launch_bounds__(256) void final_kernel(const float* __restrict__ scores,
                                                    const float* __restrict__ wers,
                                                    float* __restrict__ out) {
    __shared__ float acc[B_];
    const int wid  = threadIdx.x >> 5;
    const int lane = threadIdx.x & 31;

    for (int b = wid; b < B_; b += 8) {
        float v[4], w[4];
#pragma unroll
        for (int k = 0; k < 4; k++) {
            int p   = lane + 32 * k;
            bool ok = p < P_;
            v[k] = ok ? scores[b * P_ + p] : -1e30f;
            w[k] = ok ? wers[b * P_ + p] : 0.0f;
        }
        float m = fmaxf(fmaxf(v[0], v[1]), fmaxf(v[2], v[3]));
        m = wave_max(m);
        float num = 0.0f, den = 0.0f;
#pragma unroll
        for (int k = 0; k < 4; k++) {
            int p = lane + 32 * k;
            float e = (p < P_) ? __expf(v[k] - m) : 0.0f;
            num += e * w[k];
            den += e;
        }
        num = wave_sum(num);
        den = wave_sum(den);
        if (lane == 0) acc[b] = num / den;
    }
    __syncthreads();
    if (threadIdx.x == 0) {
        float s = 0.0f;
#pragma unroll
        for (int i = 0; i < B_; i++) s += acc[i];
        out[0] = s;
    }
}

// ---------------------------------------------------------------------------
extern "C" void kernel_launch(void* const* d_in, const int* in_sizes, int n_in,
                              void* d_out, int out_size, void* d_ws, size_t ws_size,
                              hipStream_t stream) {
    const float* emissions = (const float*)d_in[0];   // (B,T,V) f32
    const int*   elens     = (const int*)d_in[1];     // (B,)
    const int*   labels    = (const int*)d_in[2];     // (B,L)
    const int*   llens     = (const int*)d_in[3];     // (B,)
    const int*   paths     = (const int*)d_in[4];     // (B,P,T)
    float*       out       = (float*)d_out;

    float* lse    = (float*)d_ws;            // B*T
    float* scores = lse + B_ * T_;           // B*P
    float* wers   = scores + B_ * P_;        // B*P

    lse_kernel<<<B_ * T_, 256, 0, stream>>>(emissions, lse);
    score_kernel<<<(B_ * P_) / 8, 256, 0, stream>>>(emissions, elens, paths, lse, scores);
    edit_kernel<<<B_ * ((P_ + 7) / 8), 256, 0, stream>>>(paths, elens, labels, llens, wers);
    final_kernel<<<1, 256, 0, stream>>>(scores, wers, out);
}